// GCN_GraphClassification_25572235280972
// MI455X (gfx1250) — compile-verified
//
#include <hip/hip_runtime.h>

typedef float v2f __attribute__((ext_vector_type(2)));
typedef float v8f __attribute__((ext_vector_type(8)));
typedef unsigned int u32x4 __attribute__((ext_vector_type(4)));
typedef int i32x4 __attribute__((ext_vector_type(4)));
typedef int i32x8 __attribute__((ext_vector_type(8)));

#define HDIM   64
#define GRAPHS 512
#define CLS    10
#define BN_EPS 1e-5f

__device__ __forceinline__ void atomAddF(float* p, float v) {
  // native global_atomic_add_f32 (no CAS loop)
  unsafeAtomicAdd(p, v);
}

// ---------------- Tensor Data Mover: 2-D tile Global -> LDS ----------------
// D# per CDNA5 ISA. group0: count=1 | lds_addr |
// global_addr(57b) | type=2. group1: data_size=4B, tensor_dim0/1, tile_dim0/1,
// tensor_dim0_stride. Rows beyond tensor_dim1 read as zero (free padding).
// This toolchain's builtin takes 6 args (g0, g1, g2, g3, g4, cpol).
__device__ __forceinline__ void tdm_load_2d(const float* gsrc, unsigned lds_byte,
                                            unsigned dim0, unsigned dim1,
                                            unsigned tile0, unsigned tile1,
                                            unsigned stride0) {
  unsigned long long ga = (unsigned long long)(uintptr_t)gsrc;
  u32x4 g0;
  g0[0] = 1u;                                            // count=1, user D#
  g0[1] = lds_byte;                                      // lds_addr
  g0[2] = (unsigned)(ga & 0xFFFFFFFFu);                  // global_addr[31:0]
  g0[3] = (unsigned)((ga >> 32) & 0x01FFFFFFu) | (2u << 30);  // addr[56:32], type=2

  i32x8 g1;
  g1[0] = (int)(2u << 16);                               // data_size = 4 bytes
  g1[1] = (int)((dim0 & 0xFFFFu) << 16);                 // tensor_dim0[15:0]
  g1[2] = (int)((dim0 >> 16) | ((dim1 & 0xFFFFu) << 16));// dim0[31:16] | dim1[15:0]
  g1[3] = (int)((dim1 >> 16) | (tile0 << 16));           // dim1[31:16] | tile_dim0
  g1[4] = (int)(tile1 & 0xFFFFu);                        // tile_dim1, tile_dim2=0
  g1[5] = (int)stride0;                                  // tensor_dim0_stride[31:0]
  g1[6] = 0;
  g1[7] = 0;
  i32x4 z4 = {0, 0, 0, 0};
  i32x8 z8 = {0, 0, 0, 0, 0, 0, 0, 0};
  __builtin_amdgcn_tensor_load_to_lds(g0, g1, z4, z4, z8, 0);
}

// ---------------- elementwise / setup kernels ----------------

__global__ void k_fill(float* __restrict__ p, float val, int n) {
  int i = blockIdx.x * blockDim.x + threadIdx.x;
  if (i < n) p[i] = val;
}

__global__ void k_deg(const int* __restrict__ ei, float* __restrict__ deg, int E) {
  int e = blockIdx.x * blockDim.x + threadIdx.x;
  if (e < E) atomAddF(&deg[ei[E + e]], 1.0f);
}

__global__ void k_dinv(float* __restrict__ deg, float* __restrict__ selfc, int n) {
  int i = blockIdx.x * blockDim.x + threadIdx.x;
  if (i < n) {
    float d = rsqrtf(deg[i]);
    deg[i]   = d;        // deg buffer becomes dinv
    selfc[i] = d * d;    // self-loop coefficient
  }
}

__global__ void k_coef(const int* __restrict__ ei, const float* __restrict__ dinv,
                       float* __restrict__ coef, int E) {
  int e = blockIdx.x * blockDim.x + threadIdx.x;
  if (e < E) coef[e] = dinv[ei[e]] * dinv[ei[E + e]];
}

// ---------------- WMMA fp32 GEMM: [M x K] @ [K x 64] -> [M x 64] ----------------
// 128 threads = 4 wave32. Each wave: 16 rows x 64 cols (4 C-tiles of 16x16).
// A slice (16xK per wave) is DMA'd to LDS by the Tensor Data Mover; B is staged
// cooperatively into LDS *pre-paired* ({B[2r][c],B[2r+1][c]} as float2) so every
// B fragment is a single aligned ds_load_b64 straight into a WMMA operand pair.
// Fragment layouts per CDNA5 ISA:
//   A 16x4 : lanes 0-15 M=lane {v0=K,v1=K+1}; lanes 16-31 {v0=K+2,v1=K+3}
//   B 4x16 : v0 = {B[K][n] | B[K+2][n]}, v1 = {B[K+1][n] | B[K+3][n]}
//   C 16x16: VGPR v, lanes 0-15 M=v, lanes 16-31 M=v+8
// NOTE: `out` must be padded to a multiple of 64 rows (stores are unguarded).

template <int K>
__global__ __launch_bounds__(128) void gemm_wmma(const float* __restrict__ in,
                                                 const float* __restrict__ w,
                                                 float* __restrict__ out, int M) {
  __shared__ float Asmem[64 * K];
  __shared__ v2f   Bp[(K / 2) * HDIM];   // Bp[r*64+c] = {B[2r][c], B[2r+1][c]}

  const int tid  = threadIdx.x;
  const int wave = tid >> 5;
  const int lane = tid & 31;
  const int hi   = lane >> 4;   // half-wave select
  const int lo   = lane & 15;
  const int rowbase   = blockIdx.x * 64;
  const int row_start = rowbase + wave * 16;

  // per-wave TDM: own 16xK A slice; rows past M read back as zero
  int rows_avail = M - row_start;
  if (rows_avail < 0) rows_avail = 0;
  tdm_load_2d(in + (size_t)row_start * K,
              (unsigned)(uintptr_t)&Asmem[wave * 16 * K],
              (unsigned)K, (unsigned)rows_avail,
              (unsigned)K, 16u, (unsigned)K);

  // cooperative B staging, interleaving row pairs into float2 entries
  for (int idx = tid; idx < (K / 2) * HDIM; idx += 128) {
    int r = idx >> 6;          // row pair
    int c = idx & (HDIM - 1);  // column
    v2f val;
    val.x = w[(2 * r)     * HDIM + c];
    val.y = w[(2 * r + 1) * HDIM + c];
    Bp[idx] = val;
  }

  // The TDM is the only writer of Asmem that the compiler cannot see. This
  // never-taken (but unprovable) store keeps Asmem "may-be-written" so the
  // ds_loads below are not folded away. Costs one scalar compare.
  if (M < 0) Asmem[tid] = 0.0f;

  __builtin_amdgcn_s_wait_tensorcnt(0);
  __syncthreads();   // LDS fence: publish Bp (and A slices) to all waves

  v8f acc[4] = {};
  const int arow = wave * 16 + lo;
  const int hi2  = hi * 2;

#pragma unroll 4
  for (int k = 0; k < K; k += 4) {
    v2f a = *(const v2f*)&Asmem[arow * K + k + hi2];     // ds_load_b64
    const int bbase = ((k >> 1) + hi) * HDIM + lo;       // row-pair base
#pragma unroll
    for (int t = 0; t < 4; ++t) {
      v2f b = Bp[bbase + t * 16];                        // ds_load_b64
      acc[t] = __builtin_amdgcn_wmma_f32_16x16x4_f32(
          /*neg_a=*/false, a, /*neg_b=*/false, b,
          /*c_mod=*/(short)0, acc[t], /*reuse_a=*/false, /*reuse_b=*/false);
    }
  }

  // unguarded immediate-offset stores (out padded to 64-row multiple)
  float* po = out + (size_t)(row_start + hi * 8) * HDIM + lo;
#pragma unroll
  for (int t = 0; t < 4; ++t) {
#pragma unroll
    for (int v = 0; v < 8; ++v) {
      po[v * HDIM + t * 16] = acc[t][v];
    }
  }
}

// ---------------- edge scatter: agg[dst] += h[src] * coef ----------------

__global__ void k_scatter(const int* __restrict__ ei, const float* __restrict__ hw,
                          const float* __restrict__ coef, float* __restrict__ agg, int E) {
  long long idx = (long long)blockIdx.x * blockDim.x + threadIdx.x;
  if (idx >= (long long)E * 16) return;
  int e = (int)(idx >> 4);
  int q = (int)(idx & 15);
  int s = ei[e];
  int d = ei[E + e];
  float c = coef[e];
  float4 v = *(const float4*)(hw + (size_t)s * HDIM + q * 4);
  float* a = agg + (size_t)d * HDIM + q * 4;
  atomAddF(a + 0, v.x * c);
  atomAddF(a + 1, v.y * c);
  atomAddF(a + 2, v.z * c);
  atomAddF(a + 3, v.w * c);
}

// ---------------- fused self-loop + bias + BN(eval) + ReLU ----------------

__global__ void k_epilogue(const float* __restrict__ agg, const float* __restrict__ hw,
                           const float* __restrict__ selfc, const float* __restrict__ bconv,
                           const float* __restrict__ gamma, const float* __restrict__ beta,
                           const float* __restrict__ mean, const float* __restrict__ var,
                           float* __restrict__ hout, long long total) {
  long long idx = (long long)blockIdx.x * blockDim.x + threadIdx.x;
  if (idx >= total) return;
  int f = (int)(idx & (HDIM - 1));
  long long n = idx >> 6;
  float sc = gamma[f] * rsqrtf(var[f] + BN_EPS);
  float sh = beta[f] - mean[f] * sc;
  float val = agg[idx] + hw[idx] * selfc[n] + bconv[f];
  val = val * sc + sh;
  hout[idx] = fmaxf(val, 0.0f);
}

// ---------------- global mean pool ----------------

__global__ void k_pool(const float* __restrict__ h, const int* __restrict__ batch,
                       float* __restrict__ pooled, float* __restrict__ counts, int N) {
  long long idx = (long long)blockIdx.x * blockDim.x + threadIdx.x;
  if (idx >= (long long)N * 16) return;
  int n = (int)(idx >> 4);
  int q = (int)(idx & 15);
  int g = batch[n];
  float4 v = *(const float4*)(h + (size_t)n * HDIM + q * 4);
  float* p = pooled + (size_t)g * HDIM + q * 4;
  atomAddF(p + 0, v.x);
  atomAddF(p + 1, v.y);
  atomAddF(p + 2, v.z);
  atomAddF(p + 3, v.w);
  if (q == 0) atomAddF(&counts[g], 1.0f);
}

__global__ void k_pool_fin(float* __restrict__ pooled, const float* __restrict__ counts, int total) {
  int idx = blockIdx.x * blockDim.x + threadIdx.x;
  if (idx < total) pooled[idx] /= fmaxf(counts[idx / HDIM], 1.0f);
}

// ---------------- classifier tail ----------------

__global__ void k_bias_relu(float* __restrict__ h, const float* __restrict__ b, int total) {
  int idx = blockIdx.x * blockDim.x + threadIdx.x;
  if (idx < total) h[idx] = fmaxf(h[idx] + b[idx & (HDIM - 1)], 0.0f);
}

__global__ void k_mlp2(const float* __restrict__ h1, const float* __restrict__ wc2,
                       const float* __restrict__ bc2, float* __restrict__ out, int total) {
  int idx = blockIdx.x * blockDim.x + threadIdx.x;
  if (idx >= total) return;
  int g = idx / CLS;
  int c = idx % CLS;
  float s = bc2[c];
#pragma unroll 8
  for (int j = 0; j < HDIM; ++j) s += h1[g * HDIM + j] * wc2[j * CLS + c];
  out[idx] = s;
}

// ---------------- launch ----------------

static inline int cdiv_ll(long long a, long long b) { return (int)((a + b - 1) / b); }

extern "C" void kernel_launch(void* const* d_in, const int* in_sizes, int n_in,
                              void* d_out, int out_size, void* d_ws, size_t ws_size,
                              hipStream_t stream) {
  const float* x     = (const float*)d_in[0];
  const int*   ei    = (const int*)d_in[1];
  const int*   batch = (const int*)d_in[2];
  const float* w0    = (const float*)d_in[3];
  const float* w12   = (const float*)d_in[4];
  const float* bconv = (const float*)d_in[5];
  const float* gamma = (const float*)d_in[6];
  const float* beta  = (const float*)d_in[7];
  const float* mean  = (const float*)d_in[8];
  const float* var   = (const float*)d_in[9];
  const float* wc1   = (const float*)d_in[10];
  const float* bc1   = (const float*)d_in[11];
  const float* wc2   = (const float*)d_in[12];
  const float* bc2   = (const float*)d_in[13];
  float* out = (float*)d_out;

  const int N = in_sizes[2];          // batch vector has one entry per node
  const int E = in_sizes[1] / 2;      // edge_index is [2,E]
  const int M64 = ((N + 63) / 64) * 64;

  // workspace layout (floats)
  float* ws     = (float*)d_ws;
  float* deg    = ws;                               // N   (becomes dinv)
  float* selfc  = deg + N;                          // N
  float* coef   = selfc + N;                        // E
  float* hw     = coef + E;                         // M64*H (GEMM output, padded)
  float* agg    = hw + (size_t)M64 * HDIM;          // M64*H
  float* hcur   = agg + (size_t)M64 * HDIM;         // M64*H (layer activations)
  float* pooled = hcur + (size_t)M64 * HDIM;        // G*H
  float* counts = pooled + GRAPHS * HDIM;           // G
  float* h1     = counts + GRAPHS;                  // G*H (G multiple of 64)

  // --- GCN normalization coefficients ---
  k_fill<<<cdiv_ll(N, 256), 256, 0, stream>>>(deg, 1.0f, N);
  k_deg<<<cdiv_ll(E, 256), 256, 0, stream>>>(ei, deg, E);
  k_dinv<<<cdiv_ll(N, 256), 256, 0, stream>>>(deg, selfc, N);
  k_coef<<<cdiv_ll(E, 256), 256, 0, stream>>>(ei, deg, coef, E);

  // --- 3 GCN conv layers ---
  for (int layer = 0; layer < 3; ++layer) {
    const float* hin = (layer == 0) ? x : hcur;
    const float* w   = (layer == 0) ? w0 : (w12 + (size_t)(layer - 1) * HDIM * HDIM);
    if (layer == 0)
      gemm_wmma<128><<<cdiv_ll(N, 64), 128, 0, stream>>>(hin, w, hw, N);
    else
      gemm_wmma<64><<<cdiv_ll(N, 64), 128, 0, stream>>>(hin, w, hw, N);

    k_fill<<<cdiv_ll((long long)N * HDIM, 256), 256, 0, stream>>>(agg, 0.0f, N * HDIM);
    k_scatter<<<cdiv_ll((long long)E * 16, 256), 256, 0, stream>>>(ei, hw, coef, agg, E);
    k_epilogue<<<cdiv_ll((long long)N * HDIM, 256), 256, 0, stream>>>(
        agg, hw, selfc, bconv + layer * HDIM, gamma + layer * HDIM, beta + layer * HDIM,
        mean + layer * HDIM, var + layer * HDIM, hcur, (long long)N * HDIM);
  }

  // --- global mean pool (pooled and counts contiguous: zero together) ---
  k_fill<<<cdiv_ll(GRAPHS * HDIM + GRAPHS, 256), 256, 0, stream>>>(pooled, 0.0f,
                                                                   GRAPHS * HDIM + GRAPHS);
  k_pool<<<cdiv_ll((long long)N * 16, 256), 256, 0, stream>>>(hcur, batch, pooled, counts, N);
  k_pool_fin<<<cdiv_ll(GRAPHS * HDIM, 256), 256, 0, stream>>>(pooled, counts, GRAPHS * HDIM);

  // --- classifier: WMMA for the 512x64x64 layer, tiny kernel for 64x10 ---
  gemm_wmma<64><<<cdiv_ll(GRAPHS, 64), 128, 0, stream>>>(pooled, wc1, h1, GRAPHS);
  k_bias_relu<<<cdiv_ll(GRAPHS * HDIM, 256), 256, 0, stream>>>(h1, bc1, GRAPHS * HDIM);
  k_mlp2<<<cdiv_ll(GRAPHS * CLS, 256), 256, 0, stream>>>(h1, wc2, bc2, out, GRAPHS * CLS);
}